// RANSAC_36807869726871
// MI455X (gfx1250) — compile-verified
//
#include <hip/hip_runtime.h>

typedef float v2f __attribute__((ext_vector_type(2)));
typedef float v8f __attribute__((ext_vector_type(8)));

#define BATCH 32
#define NPTS  1024
#define KDIM  12
#define NPAIR 6        // KDIM/2 : V stored as K-pairs for b64 loads
#define PS    14.0f
#define THR2  25.0f    // PIXEL_THRESHOLD^2

// workspace float offsets
#define WS_U     0
#define WS_V     (BATCH*NPTS*KDIM)          // packed: [b][pair][m][2]
#define WS_SV    (WS_V + BATCH*KDIM*NPTS)
#define WS_VALID (WS_SV + BATCH*NPTS)
#define WS_SCORE (WS_VALID + BATCH*NPTS)

// output float offsets (tuple concatenated flat: Ms, failed, inl_src, inl_tar, inl_sc)
#define OFF_M    0
#define OFF_FAIL (BATCH*9)
#define OFF_SRC  (OFF_FAIL + BATCH)
#define OFF_TAR  (OFF_SRC + BATCH*NPTS*2)
#define OFF_SC   (OFF_TAR + BATCH*NPTS*2)

// ---------------------------------------------------------------------------
// Kernel 1: build rank-12 factors.  err2[n,m] = U[n,:] . V[:,m]
// U row-major (12/pt); V packed as K-pairs: Vp[(pair*NPTS + m)*2 + {0,1}]
// ---------------------------------------------------------------------------
__global__ void ransac_prep(const float* __restrict__ src, const float* __restrict__ tar,
                            const float* __restrict__ sc,  const float* __restrict__ rsc,
                            const float* __restrict__ rip, float* __restrict__ ws) {
    int idx = blockIdx.x * blockDim.x + threadIdx.x;   // b*NPTS + n
    int b = idx >> 10, n = idx & (NPTS - 1);
    float sx0 = src[idx*2], sy0 = src[idx*2+1];
    float tx0 = tar[idx*2], ty0 = tar[idx*2+1];
    float score = sc[idx], scale = rsc[idx];
    float cc = rip[idx*2], ss = rip[idx*2+1];
    bool valid = (sx0 != -1.0f);
    float sx = sx0*PS, sy = sy0*PS, tx = tx0*PS, ty = ty0*PS;
    float a = scale*cc, bb = scale*ss;
    float px = tx - (a*sx - bb*sy);
    float py = ty - (bb*sx + a*sy);

    float* U = ws + WS_U + (size_t)idx * KDIM;
    U[0]  = a*a + bb*bb;
    U[1]  = 2.0f*(a*px + bb*py);
    U[2]  = 2.0f*(a*py - bb*px);
    U[3]  = px*px + py*py;
    U[4]  = -2.0f*px;
    U[5]  = -2.0f*py;
    U[6]  = 1.0f;
    U[7]  = -2.0f*a;
    U[8]  =  2.0f*bb;
    U[9]  = -2.0f*bb;
    U[10] = -2.0f*a;
    U[11] = 0.0f;

    // V features, stored as adjacent (k, k+1) pairs per m for b64 B-operand loads
    float* Vp = ws + WS_V + (size_t)b * NPAIR * NPTS * 2;
    v2f p0 = { sx*sx + sy*sy, sx };        // k0,k1
    v2f p1 = { sy,            1.0f };      // k2,k3
    v2f p2 = { tx,            ty };        // k4,k5
    v2f p3 = { tx*tx + ty*ty, sx*tx };     // k6,k7
    v2f p4 = { sy*tx,         sx*ty };     // k8,k9
    v2f p5 = { sy*ty,         0.0f };      // k10,k11
    *(v2f*)(Vp + (size_t)(0*NPTS + n)*2) = p0;
    *(v2f*)(Vp + (size_t)(1*NPTS + n)*2) = p1;
    *(v2f*)(Vp + (size_t)(2*NPTS + n)*2) = p2;
    *(v2f*)(Vp + (size_t)(3*NPTS + n)*2) = p3;
    *(v2f*)(Vp + (size_t)(4*NPTS + n)*2) = p4;
    *(v2f*)(Vp + (size_t)(5*NPTS + n)*2) = p5;

    ws[WS_SV    + idx] = valid ? score : 0.0f;   // fold valid[m] into weight
    ws[WS_VALID + idx] = valid ? 1.0f : 0.0f;
}

// ---------------------------------------------------------------------------
// Kernel 2: err2 tiles via V_WMMA_F32_16X16X4_F32.
// Block = 256 threads = 8 waves; each wave owns TWO 16-row strips (32 rows),
// sharing the B operands between both accumulator chains (6 WMMAs / 4 loads).
// ---------------------------------------------------------------------------
__global__ void ransac_score(float* __restrict__ ws) {
    const int b       = blockIdx.y;
    const int w       = threadIdx.x >> 5;
    const int lane    = threadIdx.x & 31;
    const int laneLow = lane & 15;
    const int laneHi  = lane >> 4;            // 0: K0/K1 half, 1: K2/K3 half
    const int n0      = blockIdx.x * 256 + w * 32;

    const float* Ub  = ws + WS_U    + (size_t)b * NPTS * KDIM;
    const float* Vpb = ws + WS_V    + (size_t)b * NPAIR * NPTS * 2;
    const float* svb = ws + WS_SV   + (size_t)b * NPTS;
    const float* vfb = ws + WS_VALID+ (size_t)b * NPTS;

    // A operands: 16x4 layout, lanes 0-15 hold K=0,1 ; lanes 16-31 hold K=2,3
    const float* Ur0 = Ub + (size_t)(n0 + laneLow)      * KDIM + 2 * laneHi;
    const float* Ur1 = Ub + (size_t)(n0 + 16 + laneLow) * KDIM + 2 * laneHi;
    v2f a0 = *(const v2f*)(Ur0 + 0), a3 = *(const v2f*)(Ur1 + 0);
    v2f a1 = *(const v2f*)(Ur0 + 4), a4 = *(const v2f*)(Ur1 + 4);
    v2f a2 = *(const v2f*)(Ur0 + 8), a5 = *(const v2f*)(Ur1 + 8);

    float rs0[8] = {0.f,0.f,0.f,0.f,0.f,0.f,0.f,0.f};
    float rs1[8] = {0.f,0.f,0.f,0.f,0.f,0.f,0.f,0.f};

    for (int t = 0; t < NPTS / 16; ++t) {
        const int col = t * 16 + laneLow;
        float svc = svb[col];

        if (t + 1 < NPTS / 16) {  // uniform branch; exercises global_prefetch_b8
            __builtin_prefetch(Vpb + (size_t)(laneHi * NPTS + col + 16) * 2, 0, 0);
        }

        // B operands: pair index p = 2g + laneHi -> one b64 per K-group
        v2f b0 = *(const v2f*)(Vpb + (size_t)((0 + laneHi) * NPTS + col) * 2);
        v2f b1 = *(const v2f*)(Vpb + (size_t)((2 + laneHi) * NPTS + col) * 2);
        v2f b2 = *(const v2f*)(Vpb + (size_t)((4 + laneHi) * NPTS + col) * 2);

        v8f c0 = {0.f,0.f,0.f,0.f,0.f,0.f,0.f,0.f};
        v8f c1 = {0.f,0.f,0.f,0.f,0.f,0.f,0.f,0.f};
        c0 = __builtin_amdgcn_wmma_f32_16x16x4_f32(false, a0, false, b0, (short)0, c0, false, false);
        c1 = __builtin_amdgcn_wmma_f32_16x16x4_f32(false, a3, false, b0, (short)0, c1, false, false);
        c0 = __builtin_amdgcn_wmma_f32_16x16x4_f32(false, a1, false, b1, (short)0, c0, false, false);
        c1 = __builtin_amdgcn_wmma_f32_16x16x4_f32(false, a4, false, b1, (short)0, c1, false, false);
        c0 = __builtin_amdgcn_wmma_f32_16x16x4_f32(false, a2, false, b2, (short)0, c0, false, false);
        c1 = __builtin_amdgcn_wmma_f32_16x16x4_f32(false, a5, false, b2, (short)0, c1, false, false);

        #pragma unroll
        for (int r = 0; r < 8; ++r) {
            int row0 = n0 + r + 8 * laneHi;            // D: VGPR r = rows r / r+8
            int row1 = row0 + 16;
            bool ok0 = (c0[r] <= THR2) && (row0 != col);  // valid[m] folded into svc
            bool ok1 = (c1[r] <= THR2) && (row1 != col);
            rs0[r] += ok0 ? svc : 0.0f;
            rs1[r] += ok1 ? svc : 0.0f;
        }
    }

    // reduce each row-sum across the 16 lanes of its half-wave
    float* sib = ws + WS_SCORE + (size_t)b * NPTS;
    #pragma unroll
    for (int r = 0; r < 8; ++r) {
        float v0 = rs0[r], v1 = rs1[r];
        v0 += __shfl_xor(v0, 1, 32);  v1 += __shfl_xor(v1, 1, 32);
        v0 += __shfl_xor(v0, 2, 32);  v1 += __shfl_xor(v1, 2, 32);
        v0 += __shfl_xor(v0, 4, 32);  v1 += __shfl_xor(v1, 4, 32);
        v0 += __shfl_xor(v0, 8, 32);  v1 += __shfl_xor(v1, 8, 32);
        if (laneLow == 0) {
            int n = n0 + r + 8 * laneHi;
            sib[n]      = (vfb[n]      != 0.0f) ? v0 : -1.0f;
            sib[n + 16] = (vfb[n + 16] != 0.0f) ? v1 : -1.0f;
        }
    }
}

// ---------------------------------------------------------------------------
// Kernel 3: per-batch argmax (first-max), M/failed, stable inlier compaction.
// One block of 1024 threads per batch.
// ---------------------------------------------------------------------------
__global__ void ransac_final(const float* __restrict__ src, const float* __restrict__ tar,
                             const float* __restrict__ sc,  const float* __restrict__ rsc,
                             const float* __restrict__ rip, const float* __restrict__ ws,
                             float* __restrict__ out) {
    __shared__ float sVal[32];
    __shared__ int   sIdx[32];
    __shared__ int   sCnt[32];
    __shared__ int   sBest;
    __shared__ float sBestVal;

    const int b = blockIdx.x;
    const int m = threadIdx.x;
    const int lane = m & 31, w = m >> 5;

    const float* sib = ws + WS_SCORE + (size_t)b * NPTS;
    float v = sib[m];
    int   ii = m;
    #pragma unroll
    for (int off = 16; off; off >>= 1) {
        float ov = __shfl_down(v, off);
        int   oi = __shfl_down(ii, off);
        if (ov > v || (ov == v && oi < ii)) { v = ov; ii = oi; }
    }
    if (lane == 0) { sVal[w] = v; sIdx[w] = ii; }
    __syncthreads();
    if (w == 0) {
        v = sVal[lane]; ii = sIdx[lane];
        #pragma unroll
        for (int off = 16; off; off >>= 1) {
            float ov = __shfl_down(v, off);
            int   oi = __shfl_down(ii, off);
            if (ov > v || (ov == v && oi < ii)) { v = ov; ii = oi; }
        }
        if (lane == 0) { sBest = ii; sBestVal = v; }
    }
    __syncthreads();
    const int best = sBest;
    const float bestVal = sBestVal;

    // best hypothesis parameters (all threads; loads hit cache)
    const int ib = b * NPTS + best;
    float scale = rsc[ib], cc = rip[ib*2], ssn = rip[ib*2+1];
    float a = scale*cc, bb = scale*ssn;
    float sxb = src[ib*2]*PS, syb = src[ib*2+1]*PS;
    float txb = tar[ib*2]*PS, tyb = tar[ib*2+1]*PS;
    float px = txb - (a*sxb - bb*syb);
    float py = tyb - (bb*sxb + a*syb);

    // this thread's point: inlier test against the best row
    const int im = b * NPTS + m;
    float sx0 = src[im*2], sy0 = src[im*2+1];
    float tx = tar[im*2]*PS, ty = tar[im*2+1]*PS;
    float sx = sx0*PS, sy = sy0*PS;
    float dx = a*sx - bb*sy + px - tx;
    float dy = bb*sx + a*sy + py - ty;
    float e2 = dx*dx + dy*dy;
    bool inl = (e2 <= THR2) && (sx0 != -1.0f) && (m != best);

    unsigned int mask = (unsigned int)__ballot(inl);
    int lp = __popc(mask & ((1u << lane) - 1u));
    if (lane == 0) sCnt[w] = __popc(mask);
    __syncthreads();
    if (m == 0) {                 // serial scan of 32 wave counts
        int s = 0;
        for (int i = 0; i < 32; ++i) { int t = sCnt[i]; sCnt[i] = s; s += t; }
    }
    // default-fill outputs (same barrier interval as the scan; disjoint targets)
    out[OFF_SRC + (size_t)im*2 + 0] = -1.0f;
    out[OFF_SRC + (size_t)im*2 + 1] = -1.0f;
    out[OFF_TAR + (size_t)im*2 + 0] = -1.0f;
    out[OFF_TAR + (size_t)im*2 + 1] = -1.0f;
    out[OFF_SC  + im] = 0.0f;
    __syncthreads();

    if (inl) {                    // stable compaction: inliers first, original order
        int p  = sCnt[w] + lp;
        int op = b * NPTS + p;
        out[OFF_SRC + (size_t)op*2 + 0] = sx0;
        out[OFF_SRC + (size_t)op*2 + 1] = sy0;
        out[OFF_TAR + (size_t)op*2 + 0] = tar[im*2];
        out[OFF_TAR + (size_t)op*2 + 1] = tar[im*2+1];
        out[OFF_SC  + op] = sc[im];
    }

    if (m == 0) {
        float* M = out + OFF_M + b * 9;
        M[0] = a;    M[1] = -bb;  M[2] = px;
        M[3] = bb;   M[4] = a;    M[5] = py;
        M[6] = 0.f;  M[7] = 0.f;  M[8] = 1.f;
        out[OFF_FAIL + b] = (bestVal == 0.0f) ? 1.0f : 0.0f;
    }
}

// ---------------------------------------------------------------------------
extern "C" void kernel_launch(void* const* d_in, const int* in_sizes, int n_in,
                              void* d_out, int out_size, void* d_ws, size_t ws_size,
                              hipStream_t stream) {
    const float* src = (const float*)d_in[0];
    const float* tar = (const float*)d_in[1];
    const float* sc  = (const float*)d_in[2];
    const float* rsc = (const float*)d_in[3];
    const float* rip = (const float*)d_in[4];
    float* ws  = (float*)d_ws;
    float* out = (float*)d_out;

    ransac_prep <<<dim3((BATCH*NPTS)/256), dim3(256), 0, stream>>>(src, tar, sc, rsc, rip, ws);
    ransac_score<<<dim3(NPTS/256, BATCH), dim3(256), 0, stream>>>(ws);
    ransac_final<<<dim3(BATCH),           dim3(1024), 0, stream>>>(src, tar, sc, rsc, rip, ws, out);
}